// GAT_2370821947941
// MI455X (gfx1250) — compile-verified
//
#include <hip/hip_runtime.h>
#include <cstdint>
#include <cstddef>

typedef float v2f __attribute__((ext_vector_type(2)));
typedef float v8f __attribute__((ext_vector_type(8)));
typedef int async_v2i __attribute__((vector_size(8)));   // pointee type of async b64 builtin
#define GLOBAL_AS __attribute__((address_space(1)))
#define LDS_AS __attribute__((address_space(3)))

#ifndef __has_builtin
#define __has_builtin(x) 0
#endif
#if __has_builtin(__builtin_amdgcn_global_load_async_to_lds_b64) && \
    __has_builtin(__builtin_amdgcn_s_wait_asynccnt)
#define USE_ASYNC_LDS 1
#else
#define USE_ASYNC_LDS 0
#endif

// Monotonic float<->int order mapping for atomic float max via atomicMax(int).
__device__ __forceinline__ int ord_enc(float v) {
  int i = __float_as_int(v);
  return i >= 0 ? i : (i ^ 0x7fffffff);
}
__device__ __forceinline__ float ord_dec(int i) {
  return __int_as_float(i >= 0 ? i : (i ^ 0x7fffffff));
}

__global__ void fill_u32_k(uint32_t* __restrict__ p, uint32_t v, size_t n) {
  size_t i = (size_t)blockIdx.x * blockDim.x + threadIdx.x;
  if (i < n) p[i] = v;
}

// C[M,Nc] = A[M,K] * B[K,Nc]; M%16==0, Nc%16==0, K%4==0, K<=128.
// Block = 8 waves -> 128 rows x 16 cols. B panel [K x 16] staged in LDS once
// per block (gfx1250 async global->LDS, ASYNCcnt-tracked), then each wave32
// computes one 16x16 tile with V_WMMA_F32_16X16X4_F32.
// A 16x4 layout: lanes 0-15 hold M=lane {K=k,k+1}, lanes 16-31 {K=k+2,k+3}.
// C/D layout: VGPR r -> row (r + 8*(lane>>4)), col = lane&15.
__global__ void __launch_bounds__(256)
gemm_wmma_f32_lds_k(const float* __restrict__ A, const float* __restrict__ B,
                    float* __restrict__ C, int M, int Nc, int K) {
  __shared__ float Bs[2048];  // K<=128 rows x 16 cols
  int ntile = Nc >> 4;
  int mg = blockIdx.x / ntile;
  int tn = blockIdx.x - mg * ntile;
  int col0 = tn << 4;
  int t = threadIdx.x;

  // Cooperative load of B panel [K x 16] into LDS (2 floats per thread/pass).
  for (int e = t * 2; e < K * 16; e += 512) {
    int k = e >> 4, c = e & 15;
    const float* g = B + (size_t)k * Nc + col0 + c;
#if USE_ASYNC_LDS
    GLOBAL_AS async_v2i* gp =
        (GLOBAL_AS async_v2i*)(GLOBAL_AS void*)(void*)const_cast<float*>(g);
    LDS_AS async_v2i* lp = (LDS_AS async_v2i*)(LDS_AS void*)(void*)&Bs[e];
    __builtin_amdgcn_global_load_async_to_lds_b64(gp, lp, 0, 0);
#else
    Bs[e] = g[0];
    Bs[e + 1] = g[1];
#endif
  }
#if USE_ASYNC_LDS
  __builtin_amdgcn_s_wait_asynccnt(0);
#endif
  __syncthreads();

  int wave = t >> 5, lane = t & 31;
  int tm = mg * 8 + wave;
  if (tm * 16 < M) {                      // wave-uniform guard: EXEC all-ones
    int row = tm * 16 + (lane & 15);
    int cl = lane & 15;
    int khi = (lane >> 4) << 1;           // 0 for lanes 0-15, 2 for 16-31
    const float* Ar = A + (size_t)row * K;
    v8f acc = {};
    for (int k = 0; k < K; k += 4) {
      v2f a, b;
      a.x = Ar[k + khi];
      a.y = Ar[k + khi + 1];
      b.x = Bs[(k + khi) * 16 + cl];
      b.y = Bs[(k + khi + 1) * 16 + cl];
      acc = __builtin_amdgcn_wmma_f32_16x16x4_f32(false, a, false, b,
                                                  (short)0, acc, false, false);
    }
    int mb = tm * 16 + (lane >> 4) * 8;
#pragma unroll
    for (int r = 0; r < 8; ++r)
      C[(size_t)(mb + r) * Nc + col0 + cl] = acc[r];
  }
}

// asrc[n,h] = sum_d xh[n,h,d]*a_src[h,d]; same for adst.
__global__ void node_logits_k(const float* __restrict__ xh, const float* __restrict__ a_src,
                              const float* __restrict__ a_dst, float* __restrict__ asrc,
                              float* __restrict__ adst, long NH, int H, int D) {
  long i = (long)blockIdx.x * blockDim.x + threadIdx.x;   // i = n*H + h
  if (i >= NH) return;
  int h = (int)(i % H);
  const float* xr = xh + (size_t)i * D;
  const float* as = a_src + (size_t)h * D;
  const float* ad = a_dst + (size_t)h * D;
  float s1 = 0.f, s2 = 0.f;
  for (int d = 0; d < D; ++d) { float v = xr[d]; s1 += v * as[d]; s2 += v * ad[d]; }
  asrc[i] = s1;
  adst[i] = s2;
}

template <int H>
__global__ void edge_max_k(const int* __restrict__ ei, long E, long N,
                           const float* __restrict__ asrc, const float* __restrict__ adst,
                           int* __restrict__ emax) {
  long e = (long)blockIdx.x * blockDim.x + threadIdx.x;
  if (e >= E + N) return;
  int s = (e < E) ? ei[e] : (int)(e - E);
  int d = (e < E) ? ei[E + e] : (int)(e - E);
#pragma unroll
  for (int h = 0; h < H; ++h) {
    float v = asrc[(size_t)s * H + h] + adst[(size_t)d * H + h];
    v = v > 0.f ? v : 0.2f * v;                 // LeakyReLU(0.2)
    atomicMax(&emax[(size_t)d * H + h], ord_enc(v));
  }
}

template <int H>
__global__ void edge_expsum_k(const int* __restrict__ ei, long E, long N,
                              const float* __restrict__ asrc, const float* __restrict__ adst,
                              const int* __restrict__ emax, float* __restrict__ den) {
  long e = (long)blockIdx.x * blockDim.x + threadIdx.x;
  if (e >= E + N) return;
  int s = (e < E) ? ei[e] : (int)(e - E);
  int d = (e < E) ? ei[E + e] : (int)(e - E);
#pragma unroll
  for (int h = 0; h < H; ++h) {
    float v = asrc[(size_t)s * H + h] + adst[(size_t)d * H + h];
    v = v > 0.f ? v : 0.2f * v;
    float m = ord_dec(emax[(size_t)d * H + h]);
    atomicAdd(&den[(size_t)d * H + h], __expf(v - m));
  }
}

// One wave32 per edge: gather xh[src] (coalesced), scatter-add alpha*xh into agg[dst].
template <int H, int D>
__global__ void edge_agg_k(const int* __restrict__ ei, long E, long N,
                           const float* __restrict__ asrc, const float* __restrict__ adst,
                           const int* __restrict__ emax, const float* __restrict__ den,
                           const float* __restrict__ xh, float* __restrict__ agg) {
  long wv = ((long)blockIdx.x * blockDim.x + threadIdx.x) >> 5;
  int lane = threadIdx.x & 31;
  if (wv >= E + N) return;
  int s = (wv < E) ? ei[wv] : (int)(wv - E);
  int d = (wv < E) ? ei[E + wv] : (int)(wv - E);
  const float* xs = xh + (size_t)s * (H * D);
  float* ag = agg + (size_t)d * (H * D);
  __builtin_prefetch(xs, 0, 1);                 // global_prefetch_b8 on source row
#pragma unroll
  for (int j0 = 0; j0 < H * D; j0 += 32) {
    int j = j0 + lane;
    int h = j / D;                              // D is a power of two -> shift
    float v = asrc[(size_t)s * H + h] + adst[(size_t)d * H + h];
    v = v > 0.f ? v : 0.2f * v;
    float m = ord_dec(emax[(size_t)d * H + h]);
    float alpha = __expf(v - m) / den[(size_t)d * H + h];
    atomicAdd(&ag[j], alpha * xs[j]);
  }
}

// out[n,d] = ELU(mean_h(agg[n,h,d]) + b[d])
__global__ void finalize1_k(const float* __restrict__ agg, const float* __restrict__ b,
                            float* __restrict__ out, long N, int H, int D) {
  long i = (long)blockIdx.x * blockDim.x + threadIdx.x;
  if (i >= N * D) return;
  long n = i / D;
  int d = (int)(i - n * D);
  float s = 0.f;
  for (int h = 0; h < H; ++h) s += agg[((size_t)n * H + h) * D + d];
  s = s * (1.0f / H) + b[d];
  out[i] = s > 0.f ? s : (__expf(s) - 1.0f);
}

// logits[n,c] = mean_h(agg[n,h,c]) + b[c]; write log_softmax then logits.
template <int H, int C>
__global__ void finalize2_k(const float* __restrict__ agg, const float* __restrict__ b,
                            float* __restrict__ out, long N) {
  long n = (long)blockIdx.x * blockDim.x + threadIdx.x;
  if (n >= N) return;
  float lg[C];
#pragma unroll
  for (int c = 0; c < C; ++c) {
    float s = 0.f;
#pragma unroll
    for (int h = 0; h < H; ++h) s += agg[((size_t)n * H + h) * C + c];
    lg[c] = s * (1.0f / H) + b[c];
  }
  float mx = lg[0];
#pragma unroll
  for (int c = 1; c < C; ++c) mx = fmaxf(mx, lg[c]);
  float se = 0.f;
#pragma unroll
  for (int c = 0; c < C; ++c) se += __expf(lg[c] - mx);
  float lse = mx + __logf(se);
#pragma unroll
  for (int c = 0; c < C; ++c) {
    out[(size_t)n * C + c] = lg[c] - lse;           // log_softmax
    out[(size_t)N * C + (size_t)n * C + c] = lg[c]; // logits
  }
}

static inline unsigned nblk(long n) { return (unsigned)((n + 255) / 256); }

extern "C" void kernel_launch(void* const* d_in, const int* in_sizes, int n_in,
                              void* d_out, int out_size, void* d_ws, size_t ws_size,
                              hipStream_t stream) {
  (void)n_in; (void)out_size; (void)ws_size;
  const float* x   = (const float*)d_in[0];
  const int*   ei  = (const int*)d_in[1];
  const float* W1  = (const float*)d_in[2];
  const float* as1 = (const float*)d_in[3];
  const float* ad1 = (const float*)d_in[4];
  const float* b1  = (const float*)d_in[5];
  const float* W2  = (const float*)d_in[6];
  const float* as2 = (const float*)d_in[7];
  const float* ad2 = (const float*)d_in[8];
  const float* b2  = (const float*)d_in[9];

  const int  HD1 = in_sizes[3];            // H*D = 256
  const int  D1  = in_sizes[5];            // 32
  const int  H   = HD1 / D1;               // 8
  const int  F   = in_sizes[2] / HD1;      // 128
  const long N   = (long)in_sizes[0] / F;  // 100000
  const long E   = (long)in_sizes[1] / 2;  // 1.6M
  const int  C   = in_sizes[9];            // 16
  const int  HC  = H * C;                  // 128
  const long EN  = E + N;

  // Workspace carve (floats); xh/agg regions reused across layers.
  float* ws   = (float*)d_ws;
  float* xh   = ws;                          // N*HD1 (layer2: N*HC)
  float* agg  = xh  + (size_t)N * HD1;       // N*HD1 (layer2: N*HC)
  float* asrc = agg + (size_t)N * HD1;       // N*H
  float* adst = asrc + (size_t)N * H;        // N*H
  int*   emax = (int*)(adst + (size_t)N * H);// N*H (ordered-int space)
  float* den  = (float*)(emax + (size_t)N * H); // N*H
  float* hbuf = den + (size_t)N * H;         // N*D1

  const int B = 256;
  const long mgroups = (N / 16 + 7) / 8;     // 8 M-tiles (128 rows) per block

  // ---------------- Layer 1 ----------------
  fill_u32_k<<<nblk(N * HD1), B, 0, stream>>>((uint32_t*)agg, 0u, (size_t)N * HD1);
  fill_u32_k<<<nblk(N * H), B, 0, stream>>>((uint32_t*)emax, 0x80000000u, (size_t)N * H);
  fill_u32_k<<<nblk(N * H), B, 0, stream>>>((uint32_t*)den, 0u, (size_t)N * H);

  gemm_wmma_f32_lds_k<<<(unsigned)(mgroups * (HD1 / 16)), B, 0, stream>>>(
      x, W1, xh, (int)N, HD1, F);
  node_logits_k<<<nblk(N * H), B, 0, stream>>>(xh, as1, ad1, asrc, adst, N * H, H, D1);
  edge_max_k<8><<<nblk(EN), B, 0, stream>>>(ei, E, N, asrc, adst, emax);
  edge_expsum_k<8><<<nblk(EN), B, 0, stream>>>(ei, E, N, asrc, adst, emax, den);
  edge_agg_k<8, 32><<<nblk(EN * 32), B, 0, stream>>>(ei, E, N, asrc, adst, emax, den, xh, agg);
  finalize1_k<<<nblk(N * D1), B, 0, stream>>>(agg, b1, hbuf, N, H, D1);

  // ---------------- Layer 2 ----------------
  fill_u32_k<<<nblk(N * HC), B, 0, stream>>>((uint32_t*)agg, 0u, (size_t)N * HC);
  fill_u32_k<<<nblk(N * H), B, 0, stream>>>((uint32_t*)emax, 0x80000000u, (size_t)N * H);
  fill_u32_k<<<nblk(N * H), B, 0, stream>>>((uint32_t*)den, 0u, (size_t)N * H);

  gemm_wmma_f32_lds_k<<<(unsigned)(mgroups * (HC / 16)), B, 0, stream>>>(
      hbuf, W2, xh, (int)N, HC, D1);
  node_logits_k<<<nblk(N * H), B, 0, stream>>>(xh, as2, ad2, asrc, adst, N * H, H, C);
  edge_max_k<8><<<nblk(EN), B, 0, stream>>>(ei, E, N, asrc, adst, emax);
  edge_expsum_k<8><<<nblk(EN), B, 0, stream>>>(ei, E, N, asrc, adst, emax, den);
  edge_agg_k<8, 16><<<nblk(EN * 32), B, 0, stream>>>(ei, E, N, asrc, adst, emax, den, xh, agg);
  finalize2_k<8, 16><<<nblk(N), B, 0, stream>>>(agg, b2, (float*)d_out, N);
}